// OpticalConv2dNew_65438121722465
// MI455X (gfx1250) — compile-verified
//
#include <hip/hip_runtime.h>

// Problem constants (from reference): B=512, IC=64, OC=128, K=5, H=28
#define NB   512
#define ICN  64
#define OCN  128
#define KS   5
#define H    28
#define HH   784      // 28*28, divisible by 16 -> 49 pixel-tiles per image
#define TAPS 25       // 5*5
#define KP   28       // taps padded to 7 WMMA k-steps of 4

typedef __attribute__((ext_vector_type(2))) float v2f;
typedef __attribute__((ext_vector_type(8))) float v8f;

// ---------------------------------------------------------------------------
// Kernel 1: xs[b, s] = sum_ic x[b, ic, s]   (reads coalesced over s)
// ---------------------------------------------------------------------------
__global__ __launch_bounds__(256)
void reduce_ic_kernel(const float* __restrict__ x, float* __restrict__ xs) {
    int p = blockIdx.x * 256 + threadIdx.x;     // 0 .. 512*784-1 (exact grid)
    int b = p / HH;
    int s = p - b * HH;
    const float* base = x + (size_t)b * ICN * HH + s;
    float acc = 0.0f;
#pragma unroll 8
    for (int ic = 0; ic < ICN; ++ic) {
        acc += base[(size_t)ic * HH];
    }
    xs[p] = acc;
}

// ---------------------------------------------------------------------------
// Kernel 2: Bpack[t*OCN + oc] = kernel[oc, ic=0, t]  (t<25), zeros for t=25..27
// ---------------------------------------------------------------------------
__global__ __launch_bounds__(256)
void pack_b_kernel(const float* __restrict__ w, float* __restrict__ Bp) {
    for (int i = threadIdx.x; i < KP * OCN; i += 256) {
        int t  = i / OCN;
        int oc = i - t * OCN;
        Bp[i] = (t < TAPS) ? w[(size_t)oc * ICN * TAPS + t] : 0.0f;
    }
}

// ---------------------------------------------------------------------------
// Kernel 3: implicit-GEMM circular conv with V_WMMA_F32_16X16X4_F32
//   grid = (3136, 8), block = 256 (8 wave32 waves)
//   blockIdx.y -> 16-OC tile (B staged in LDS, shared by all 8 waves)
//   wave       -> one 16-pixel M-tile (8 per block)
// WMMA f32 16x16x4 operand layout (ISA 7.12.2):
//   A (16x4): lanes 0-15 hold M=lane, v[0]=K0 v[1]=K1; lanes 16-31 K2/K3
//   B (4x16): lanes 0-15 hold N=lane, v[0]=K0 v[1]=K1; lanes 16-31 K2/K3
//   C/D (16x16): v[r] = row (r + 8*(lane>=16)), col = lane%16
// ---------------------------------------------------------------------------
__global__ __launch_bounds__(256)
void conv_wmma_kernel(const float* __restrict__ xs, const float* __restrict__ Bp,
                      float* __restrict__ out) {
    __shared__ float Bs[KP * 16];        // 28 taps x 16 ocs  (1.75 KB)
    __shared__ float Ts[8 * 16 * 16];    // per-wave C-transpose tiles (8 KB)

    const int octile = blockIdx.y;

    // Stage this block's B tile into LDS (all 8 waves share it).
    for (int i = threadIdx.x; i < KP * 16; i += 256) {
        int t = i >> 4;
        int n = i & 15;
        Bs[i] = Bp[t * OCN + octile * 16 + n];
    }
    __syncthreads();

    const int wave  = threadIdx.x >> 5;
    const int lane  = threadIdx.x & 31;
    const int mtile = blockIdx.x * 8 + wave;          // 0 .. 25087
    const int b     = mtile / 49;                     // image index
    const int sbase = (mtile - b * 49) * 16;          // spatial base within image

    const int hi = lane >> 4;          // half-wave: selects K pair {0,1} vs {2,3}
    const int m  = lane & 15;          // A-matrix row  = pixel within tile
    const int n  = lane & 15;          // B-matrix col  = oc within tile
    const int s  = sbase + m;
    const int ny = s / H;
    const int nx = s - ny * H;
    const float* xsb = xs + (size_t)b * HH;

    __builtin_prefetch(xsb + ny * H, 0, 0);           // global_prefetch_b8

    v8f acc = {};
#pragma unroll
    for (int ks = 0; ks < 7; ++ks) {
        const int t0 = ks * 4 + hi * 2;
        const int t1 = t0 + 1;

        // Decode circular-window source coords (clamped tap for safe address;
        // value masked to 0 for padded taps -> pure v_cndmask, no divergence).
        const int te0 = (t0 < TAPS) ? t0 : 0;
        const int te1 = (t1 < TAPS) ? t1 : 0;
        int qy0 = te0 / KS, qx0 = te0 - qy0 * KS;
        int qy1 = te1 / KS, qx1 = te1 - qy1 * KS;

        int my0 = ny + 3 - qy0; my0 += (my0 < 0) ? H : 0; my0 -= (my0 >= H) ? H : 0;
        int mx0 = nx + 3 - qx0; mx0 += (mx0 < 0) ? H : 0; mx0 -= (mx0 >= H) ? H : 0;
        int my1 = ny + 3 - qy1; my1 += (my1 < 0) ? H : 0; my1 -= (my1 >= H) ? H : 0;
        int mx1 = nx + 3 - qx1; mx1 += (mx1 < 0) ? H : 0; mx1 -= (mx1 >= H) ? H : 0;

        float a0 = xsb[my0 * H + mx0];
        float a1 = xsb[my1 * H + mx1];
        a0 = (t0 < TAPS) ? a0 : 0.0f;
        a1 = (t1 < TAPS) ? a1 : 0.0f;

        v2f Av = {a0, a1};
        v2f Bv = {Bs[t0 * 16 + n], Bs[t1 * 16 + n]};

        // D = A x B + C   (fp32 WMMA; args: neg_a, A, neg_b, B, c_mod, C, reuse_a, reuse_b)
        acc = __builtin_amdgcn_wmma_f32_16x16x4_f32(
            false, Av, false, Bv, (short)0, acc, false, false);
    }

    // |.| and transpose C tile through LDS so stores coalesce per OC row.
    float* tile = &Ts[wave * 256];
#pragma unroll
    for (int r = 0; r < 8; ++r) {
        int mm = r + hi * 8;                 // pixel row of this accumulator
        tile[n * 16 + mm] = __builtin_fabsf(acc[r]);   // tile[oc][pix]
    }
    __syncthreads();

    const size_t obase = ((size_t)b * OCN + octile * 16) * HH + sbase;
#pragma unroll
    for (int i = 0; i < 8; ++i) {
        int flat = i * 32 + lane;
        int ocr  = flat >> 4;                // 0..15
        int pix  = flat & 15;                // 0..15
        out[obase + (size_t)ocr * HH + pix] = tile[ocr * 16 + pix];
    }
}

// ---------------------------------------------------------------------------
extern "C" void kernel_launch(void* const* d_in, const int* in_sizes, int n_in,
                              void* d_out, int out_size, void* d_ws, size_t ws_size,
                              hipStream_t stream) {
    const float* x = (const float*)d_in[0];   // (512, 64, 28, 28) f32
    const float* w = (const float*)d_in[1];   // (128, 64, 5, 5)   f32
    float* out = (float*)d_out;               // (512, 128, 28, 28) f32

    float* xs = (float*)d_ws;                 // 512*784 floats
    float* Bp = xs + (size_t)NB * HH;         // 28*128 floats

    reduce_ic_kernel<<<dim3((NB * HH) / 256), dim3(256), 0, stream>>>(x, xs);
    pack_b_kernel<<<dim3(1), dim3(256), 0, stream>>>(w, Bp);

    // 25088 M-tiles (16 pixels each) / 8 waves per block = 3136 blocks in x,
    // 8 OC-tiles in y.
    conv_wmma_kernel<<<dim3(3136, 8), dim3(256), 0, stream>>>(xs, Bp, out);
}